// HarrisNet_67972152426889
// MI455X (gfx1250) — compile-verified
//
#include <hip/hip_runtime.h>
#include <hip/hip_bf16.h>

typedef __attribute__((ext_vector_type(2))) float v2f;
typedef __attribute__((ext_vector_type(8))) float v8f;

#define HH 2048
#define WW 2048
#define NIMG 4
#define TILE 16
#define ALPHA 0.05f

// per-wave LDS float counts (odd strides -> fewer LDS bank conflicts)
#define XS_STRIDE 27
#define XS_SIZE (26 * 27)     // 26x26 input halo (sobel pad 1 + gauss pad 4 = 5)
#define PS_STRIDE 25
#define PS_CH (24 * 25)       // 24x24 products per channel
#define PS_SIZE (3 * PS_CH)
#define VS_STRIDE 25
#define VS_SIZE (16 * 25)     // 16x24 vertically-blurred scratch (transpose staging)
#define WAVE_LDS (XS_SIZE + PS_SIZE + VS_SIZE)
#define WAVES_PER_BLOCK 4

#define K_TARGET 8388607u     // (4*2048*2048 - 1) / 2

__device__ __forceinline__ float band9(const float g[9], int d) {
    // gn[d] if 0 <= d <= 8 else 0 (compile-time-unrolled select chain, no scratch)
    float r = 0.0f;
#pragma unroll
    for (int k = 0; k < 9; ++k) r = (d == k) ? g[k] : r;
    return r;
}

// ---------------------------------------------------------------------------
// Kernel 1: fused Sobel -> products -> separable 9x9 Gaussian (via WMMA
// banded matmuls) -> Harris response R.  One wave per 16x16 output tile.
// Interior tiles stage their halo with GLOBAL_LOAD_ASYNC_TO_LDS (ASYNCcnt).
// ---------------------------------------------------------------------------
__global__ __launch_bounds__(128) void harris_R_kernel(const float* __restrict__ x,
                                                       const float* __restrict__ gw,
                                                       float* __restrict__ R) {
    __shared__ float smem[WAVES_PER_BLOCK * WAVE_LDS];
    const int wave = threadIdx.x >> 5;
    const int lane = threadIdx.x & 31;
    const int hl = lane >> 4;   // which 16-lane half
    const int lm = lane & 15;

    float* XS = smem + wave * WAVE_LDS;
    float* PS = XS + XS_SIZE;
    float* VS = PS + PS_SIZE;

    const int tile = blockIdx.x * WAVES_PER_BLOCK + wave;
    const int n  = tile >> 14;            // tile / (128*128)
    const int tr = (tile >> 7) & 127;
    const int tc = tile & 127;
    const int r0 = tr * TILE, c0 = tc * TILE;
    const float* xi = x + (size_t)n * HH * WW;

    // 1D gaussian taps from the passed 9x9 separable kernel:
    // g2 = gn (x) gn  =>  gn[k] = g2[4][k] / sqrt(g2[4][4])
    float gk[9];
    {
        float inv = rsqrtf(gw[40]);
#pragma unroll
        for (int d = 0; d < 9; ++d) gk[d] = gw[36 + d] * inv;
    }
    // Per-lane banded Gaussian operand values for A (vertical pass) and B
    // (horizontal pass) -- identical by symmetry: gn[t - lm] gated to [0,8].
    float ga[6], gb[6];
#pragma unroll
    for (int i = 0; i < 6; ++i) {
        int t = 4 * i + 2 * hl;
        ga[i] = band9(gk, t - lm);
        gb[i] = band9(gk, t + 1 - lm);
    }

    // Stage 26x26 input halo into LDS.
    // Interior tiles: async global->LDS DMA (no VGPR round trip, ASYNCcnt).
    const bool interior = (tr >= 1) && (tr <= 126) && (tc >= 1) && (tc <= 126);
    if (interior) {
        const float* gbase = xi + (size_t)(r0 - 5) * WW + (c0 - 5);
        const unsigned xs_base = (unsigned)(uintptr_t)XS;  // flat->LDS: low 32 bits
        for (int idx = lane; idx < 26 * 26; idx += 32) {
            int rr = idx / 26, cc = idx - rr * 26;
            unsigned lds_addr = xs_base + (unsigned)((rr * XS_STRIDE + cc) * 4);
            const float* g = gbase + (size_t)rr * WW + cc;
            asm volatile("global_load_async_to_lds_b32 %0, %1, off"
                         :: "v"(lds_addr), "v"(g) : "memory");
        }
        asm volatile("s_wait_asynccnt 0x0" ::: "memory");
    } else {
        for (int idx = lane; idx < 26 * 26; idx += 32) {
            int rr = idx / 26, cc = idx - rr * 26;
            int gr = r0 - 5 + rr, gc = c0 - 5 + cc;
            float v = 0.0f;
            if (gr >= 0 && gr < HH && gc >= 0 && gc < WW) v = xi[(size_t)gr * WW + gc];
            XS[rr * XS_STRIDE + cc] = v;
        }
    }
    __syncthreads();

    // Sobel + products over 24x24 (gauss zero-pads the PRODUCTS, so mask
    // out-of-image pixels to zero even though sobel-of-padded-x exists there)
    for (int idx = lane; idx < 24 * 24; idx += 32) {
        int pr = idx / 24, pc = idx - pr * 24;
        const float* c = XS + pr * XS_STRIDE + pc;  // 3x3 top-left (center = pr+1,pc+1)
        float a00 = c[0], a01 = c[1], a02 = c[2];
        float a10 = c[XS_STRIDE], a12 = c[XS_STRIDE + 2];
        float a20 = c[2 * XS_STRIDE], a21 = c[2 * XS_STRIDE + 1], a22 = c[2 * XS_STRIDE + 2];
        float Ix = (a02 - a00) + 2.0f * (a12 - a10) + (a22 - a20);
        float Iy = (a20 - a00) + 2.0f * (a21 - a01) + (a22 - a02);
        int gr = r0 - 4 + pr, gc = c0 - 4 + pc;
        float m = (gr >= 0 && gr < HH && gc >= 0 && gc < WW) ? 1.0f : 0.0f;
        PS[0 * PS_CH + pr * PS_STRIDE + pc] = m * Ix * Ix;
        PS[1 * PS_CH + pr * PS_STRIDE + pc] = m * Iy * Iy;
        PS[2 * PS_CH + pr * PS_STRIDE + pc] = m * Ix * Iy;
    }
    __syncthreads();

    v8f Sq[3];
#pragma unroll
    for (int q = 0; q < 3; ++q) {
        const float* P = PS + q * PS_CH;
        // Vertical pass: V(16x24) = Gv(16x24) x P(24x24), as two 16x16 D tiles
        v8f dL = {0, 0, 0, 0, 0, 0, 0, 0};
        v8f dR = {0, 0, 0, 0, 0, 0, 0, 0};
#pragma unroll
        for (int i = 0; i < 6; ++i) {
            int tA = 4 * i + 2 * hl;
            v2f a;  a.x = ga[i];                         a.y = gb[i];
            v2f bL; bL.x = P[tA * PS_STRIDE + lm];       bL.y = P[(tA + 1) * PS_STRIDE + lm];
            v2f bR; bR.x = P[tA * PS_STRIDE + lm + 8];   bR.y = P[(tA + 1) * PS_STRIDE + lm + 8];
            dL = __builtin_amdgcn_wmma_f32_16x16x4_f32(false, a, false, bL, (short)0, dL, false, false);
            dR = __builtin_amdgcn_wmma_f32_16x16x4_f32(false, a, false, bR, (short)0, dR, false, false);
        }
        // D-layout -> LDS (row-major 16x24) for the A-operand transpose
#pragma unroll
        for (int j = 0; j < 8; ++j) {
            VS[(j + 8 * hl) * VS_STRIDE + lm]     = dL[j];
            VS[(j + 8 * hl) * VS_STRIDE + lm + 8] = dR[j];
        }
        __syncthreads();
        // Horizontal pass: S(16x16) = V(16x24) x Gh(24x16)
        v8f s = {0, 0, 0, 0, 0, 0, 0, 0};
#pragma unroll
        for (int i = 0; i < 6; ++i) {
            int tA = 4 * i + 2 * hl;
            v2f a; a.x = VS[lm * VS_STRIDE + tA]; a.y = VS[lm * VS_STRIDE + tA + 1];
            v2f b; b.x = ga[i];                   b.y = gb[i];
            s = __builtin_amdgcn_wmma_f32_16x16x4_f32(false, a, false, b, (short)0, s, false, false);
        }
        Sq[q] = s;
        __syncthreads();
    }

    // Harris response, written straight from the WMMA D layout
    float* Rn = R + (size_t)n * HH * WW;
#pragma unroll
    for (int j = 0; j < 8; ++j) {
        float sxx = Sq[0][j], syy = Sq[1][j], sxy = Sq[2][j];
        float trc = sxx + syy;
        float r = sxx * syy - sxy * sxy - ALPHA * trc * trc;
        Rn[(size_t)(r0 + j + 8 * hl) * WW + (c0 + lm)] = r;
    }
}

// ---------------------------------------------------------------------------
// Median via 3-pass radix selection (bits 31:21, 20:10, 9:0 of the
// order-preserving uint mapping). Deterministic integer atomics.
// ctl layout: [0..2047] bins, [2048] k_remaining, [2049] prefix,
//             [2050] median_u, [2051] median float bits
// ---------------------------------------------------------------------------
__global__ void init_ctl(unsigned* ctl) {
    int i = blockIdx.x * 256 + threadIdx.x;
    if (i < 2048) ctl[i] = 0u;
    if (i == 0) { ctl[2048] = K_TARGET; ctl[2049] = 0u; }
}

__device__ __forceinline__ unsigned f2ord(float f) {
    unsigned b = __float_as_uint(f);
    return (b & 0x80000000u) ? ~b : (b | 0x80000000u);
}

__global__ __launch_bounds__(256) void hist_kernel(const float4* __restrict__ R4,
                                                   unsigned* __restrict__ ctl,
                                                   int shift, int nbins,
                                                   int prefshift, int use_pref, int total4) {
    __shared__ unsigned lb[2048];
    for (int i = threadIdx.x; i < nbins; i += 256) lb[i] = 0u;
    __syncthreads();
    const unsigned pref = use_pref ? ctl[2049] : 0u;
    const unsigned mask = (unsigned)(nbins - 1);
    const int stride = gridDim.x * 256;
    for (int i = blockIdx.x * 256 + threadIdx.x; i < total4; i += stride) {
        __builtin_prefetch(R4 + i + stride, 0, 1);   // -> global_prefetch_b8
        float4 v = R4[i];
        unsigned u0 = f2ord(v.x), u1 = f2ord(v.y), u2 = f2ord(v.z), u3 = f2ord(v.w);
        if (!use_pref || (u0 >> prefshift) == pref) atomicAdd(&lb[(u0 >> shift) & mask], 1u);
        if (!use_pref || (u1 >> prefshift) == pref) atomicAdd(&lb[(u1 >> shift) & mask], 1u);
        if (!use_pref || (u2 >> prefshift) == pref) atomicAdd(&lb[(u2 >> shift) & mask], 1u);
        if (!use_pref || (u3 >> prefshift) == pref) atomicAdd(&lb[(u3 >> shift) & mask], 1u);
    }
    __syncthreads();
    for (int i = threadIdx.x; i < nbins; i += 256) {
        unsigned v = lb[i];
        if (v) atomicAdd(&ctl[i], v);
    }
}

__global__ __launch_bounds__(256) void scan_kernel(unsigned* ctl, int nbins, int nbits, int last) {
    __shared__ unsigned lb[2048];
    for (int i = threadIdx.x; i < nbins; i += 256) lb[i] = ctl[i];
    __syncthreads();
    if (threadIdx.x == 0) {
        unsigned k = ctl[2048];
        unsigned pre = ctl[2049];
        unsigned c = 0, bsel = (unsigned)(nbins - 1);
        for (int i = 0; i < nbins; ++i) {
            unsigned cnt = lb[i];
            if (c + cnt > k) { bsel = (unsigned)i; break; }
            c += cnt;
        }
        ctl[2048] = k - c;
        unsigned np = (pre << nbits) | bsel;
        ctl[2049] = np;
        if (last) {
            unsigned u = np;
            unsigned fb = (u & 0x80000000u) ? (u & 0x7fffffffu) : ~u;
            ctl[2050] = u;
            ctl[2051] = fb;   // median as float bits
        }
    }
    __syncthreads();
    for (int i = threadIdx.x; i < nbins; i += 256) ctl[i] = 0u;  // ready for next pass
}

// ---------------------------------------------------------------------------
// Kernel 3: fused threshold + 7x7 max-pool NMS.  16x16 tile per block with a
// 22x22 LDS halo of thresholded values (out-of-image -> -inf, like the
// reference reduce_window identity).
// ---------------------------------------------------------------------------
__global__ __launch_bounds__(256) void nms_kernel(const float* __restrict__ R,
                                                  const unsigned* __restrict__ ctl,
                                                  float* __restrict__ out) {
    __shared__ float lt[22 * 23];
    const float med = __uint_as_float(ctl[2051]);
    const int b = blockIdx.x;
    const int n = b >> 14;
    const int tr = (b >> 7) & 127, tc = b & 127;
    const int r0 = tr * 16, c0 = tc * 16;
    const float* Rn = R + (size_t)n * HH * WW;

    for (int idx = threadIdx.x; idx < 22 * 22; idx += 256) {
        int rr = idx / 22, cc = idx - rr * 22;
        int gr = r0 - 3 + rr, gc = c0 - 3 + cc;
        float v = -INFINITY;
        if (gr >= 0 && gr < HH && gc >= 0 && gc < WW) {
            float xv = Rn[(size_t)gr * WW + gc];
            v = (xv > med) ? xv : 0.0f;
        }
        lt[rr * 23 + cc] = v;
    }
    __syncthreads();

    const int tx = threadIdx.x & 15, ty = threadIdx.x >> 4;
    const float xc = lt[(ty + 3) * 23 + (tx + 3)];
    float m = -INFINITY;
#pragma unroll
    for (int dy = 0; dy < 7; ++dy)
#pragma unroll
        for (int dx = 0; dx < 7; ++dx)
            m = fmaxf(m, lt[(ty + dy) * 23 + (tx + dx)]);
    out[((size_t)n * HH + (r0 + ty)) * WW + (c0 + tx)] = (xc == m) ? xc : 0.0f;
}

// ---------------------------------------------------------------------------
extern "C" void kernel_launch(void* const* d_in, const int* in_sizes, int n_in,
                              void* d_out, int out_size, void* d_ws, size_t ws_size,
                              hipStream_t stream) {
    const float* x  = (const float*)d_in[0];
    const float* gw = (const float*)d_in[2];   // gauss_w (9x9); sobel hardcoded
    float* out = (float*)d_out;

    const int total = NIMG * HH * WW;          // 16,777,216
    float* Rbuf = (float*)d_ws;
    unsigned* ctl = (unsigned*)((char*)d_ws + (size_t)total * sizeof(float));

    // 1) fused Harris response (WMMA separable Gaussian, async LDS staging)
    harris_R_kernel<<<(NIMG * 128 * 128) / WAVES_PER_BLOCK, 128, 0, stream>>>(x, gw, Rbuf);

    // 2) exact median: 3-pass radix select (11 + 11 + 10 bits)
    init_ctl<<<8, 256, 0, stream>>>(ctl);
    hist_kernel<<<2048, 256, 0, stream>>>((const float4*)Rbuf, ctl, 21, 2048, 0,  0, total / 4);
    scan_kernel<<<1, 256, 0, stream>>>(ctl, 2048, 11, 0);
    hist_kernel<<<2048, 256, 0, stream>>>((const float4*)Rbuf, ctl, 10, 2048, 21, 1, total / 4);
    scan_kernel<<<1, 256, 0, stream>>>(ctl, 2048, 11, 0);
    hist_kernel<<<2048, 256, 0, stream>>>((const float4*)Rbuf, ctl, 0,  1024, 10, 1, total / 4);
    scan_kernel<<<1, 256, 0, stream>>>(ctl, 1024, 10, 1);

    // 3) fused threshold + 7x7 NMS
    nms_kernel<<<NIMG * 128 * 128, 256, 0, stream>>>(Rbuf, ctl, out);
}